// PCCircuit_86114094285102
// MI455X (gfx1250) — compile-verified
//
#include <hip/hip_runtime.h>
#include <math.h>

typedef float v2f __attribute__((ext_vector_type(2)));
typedef float v8f __attribute__((ext_vector_type(8)));

#define BB 1024
#define VV 512
#define KK 8

// ---------------- Leaf layer: t0[b,v,k] = logsumexp_c( leaf[b,v,c] + log_softmax(W0[v,k,:]) )
__global__ __launch_bounds__(256) void pc_leaf(const float* __restrict__ x,
                                               const float* __restrict__ W0,
                                               float* __restrict__ t0) {
  int tid = blockIdx.x * blockDim.x + threadIdx.x;  // one thread per (b,v)
  int b = tid >> 9;      // / 512
  int v = tid & 511;
  float xv = x[(size_t)b * VV + v];
  float lx = __logf(xv);
  float l1 = log1pf(-xv);
#pragma unroll
  for (int k = 0; k < KK; ++k) {
    float w0 = W0[v * 16 + k * 2 + 0];
    float w1 = W0[v * 16 + k * 2 + 1];
    float mw = fmaxf(w0, w1);
    float lse = mw + __logf(__expf(w0 - mw) + __expf(w1 - mw));
    float a = lx + (w0 - lse);
    float c = l1 + (w1 - lse);
    float m = fmaxf(a, c);
    float t = m + __logf(__expf(a - m) + __expf(c - m));
    t0[(size_t)b * (VV * KK) + v * KK + k] = t;
  }
}

// ---------------- One sum-product layer via f32 WMMA.
// t_out[b,r,k] = log( sum_m exp(p_m - pmax) * softmax(W[r])[k,m] ) + pmax
// p[i*8+j] = left_i + right_j  =>  exp(p - pmax) = e_i * f_j  (separable)
// GEMM per (r, 16-batch-tile): P(16x64) x Wsm(64x16, cols 8..15 zero) via 16x V_WMMA_F32_16X16X4_F32
__global__ __launch_bounds__(256) void pc_layer(const float* __restrict__ tin,
                                                const float* __restrict__ W,
                                                float* __restrict__ tout,
                                                int R) {
  __shared__ float Bsm[64 * 16];   // [m][n]: softmax weights transposed, n>=8 zero
  __shared__ float pmaxS[8][16];   // per-wave, per-row pmax

  const int r    = blockIdx.x;
  const int tid  = threadIdx.x;
  const int lane = tid & 31;
  const int wave = tid >> 5;
  const int n    = lane & 15;
  const bool hi  = lane >= 16;
  const int row  = blockIdx.y * 128 + wave * 16 + n;   // batch row (M index)

  // --- block-wide: softmax of W[r] (K=8 rows of 64) into LDS as B-matrix
  if (tid < 8) {
    const float* wr = W + (size_t)r * (KK * 64) + tid * 64;
    float mx = -INFINITY;
    for (int m = 0; m < 64; ++m) mx = fmaxf(mx, wr[m]);
    float s = 0.f;
    for (int m = 0; m < 64; ++m) s += __expf(wr[m] - mx);
    float inv = 1.0f / s;
    for (int m = 0; m < 64; ++m) Bsm[m * 16 + tid] = __expf(wr[m] - mx) * inv;
  } else if (tid < 16) {
    for (int m = 0; m < 64; ++m) Bsm[m * 16 + tid] = 0.f;  // zero-pad N=8..15
  }

  // --- per-lane: left/right child states for this row
  const int R2 = R * 2;
  const float* tl = tin + (size_t)row * ((size_t)R2 * KK) + (size_t)(2 * r) * KK;
  float l[8], rg[8];
#pragma unroll
  for (int i = 0; i < 8; ++i) { l[i] = tl[i]; rg[i] = tl[8 + i]; }
  float lmax = l[0], rmax = rg[0];
#pragma unroll
  for (int i = 1; i < 8; ++i) { lmax = fmaxf(lmax, l[i]); rmax = fmaxf(rmax, rg[i]); }
  if (!hi) pmaxS[wave][n] = lmax + rmax;

  __syncthreads();

  float e[8], f[8];
#pragma unroll
  for (int i = 0; i < 8; ++i) { e[i] = __expf(l[i] - lmax); f[i] = __expf(rg[i] - rmax); }

  // --- 16 chained f32 WMMAs over the 64-element product dimension.
  // A layout (16x4 f32): lanes 0-15 hold K=0 (v0), K=1 (v1); lanes 16-31 hold K=2, K=3.
  // B layout mirrors A. Chunk c covers m = 4c..4c+3; i = c>>1, j = (c&1)*4 + t.
  v8f acc = {};
  const int hoff = hi ? 2 : 0;
#pragma unroll
  for (int c = 0; c < 16; ++c) {
    const int i  = c >> 1;
    const int jb = (c & 1) * 4;
    v2f a, bv;
    a.x  = e[i] * (hi ? f[jb + 2] : f[jb + 0]);
    a.y  = e[i] * (hi ? f[jb + 3] : f[jb + 1]);
    bv.x = Bsm[(4 * c + hoff) * 16 + n];
    bv.y = Bsm[(4 * c + hoff + 1) * 16 + n];
    acc = __builtin_amdgcn_wmma_f32_16x16x4_f32(false, a, false, bv,
                                                (short)0, acc, false, false);
  }

  // --- epilogue: C layout: vgpr g, lanes 0-15 -> (M=g, N=lane); lanes 16-31 -> (M=8+g, N=lane-16)
  if (n < 8) {
    const int mbase = hi ? 8 : 0;
    const int obase = blockIdx.y * 128 + wave * 16 + mbase;
#pragma unroll
    for (int g = 0; g < 8; ++g) {
      float val = __logf(acc[g]) + pmaxS[wave][mbase + g];
      tout[(size_t)(obase + g) * ((size_t)R * KK) + (size_t)r * KK + n] = val;
    }
  }
}

// ---------------- Root: out[b] = logsumexp_k( t[b,0,k] + log_softmax(Wroot)[k] )
__global__ __launch_bounds__(256) void pc_root(const float* __restrict__ t,
                                               const float* __restrict__ Wroot,
                                               float* __restrict__ out) {
  int b = blockIdx.x * blockDim.x + threadIdx.x;
  float w[8];
  float wm = -INFINITY;
#pragma unroll
  for (int k = 0; k < 8; ++k) { w[k] = Wroot[k]; wm = fmaxf(wm, w[k]); }
  float ws = 0.f;
#pragma unroll
  for (int k = 0; k < 8; ++k) ws += __expf(w[k] - wm);
  float lse = wm + __logf(ws);
  float a[8], am = -INFINITY;
#pragma unroll
  for (int k = 0; k < 8; ++k) { a[k] = t[(size_t)b * 8 + k] + (w[k] - lse); am = fmaxf(am, a[k]); }
  float s = 0.f;
#pragma unroll
  for (int k = 0; k < 8; ++k) s += __expf(a[k] - am);
  out[b] = am + __logf(s);
}

extern "C" void kernel_launch(void* const* d_in, const int* in_sizes, int n_in,
                              void* d_out, int out_size, void* d_ws, size_t ws_size,
                              hipStream_t stream) {
  const float* x  = (const float*)d_in[0];
  const float* W0 = (const float*)d_in[1];
  const float* Wl[9];
  for (int i = 0; i < 9; ++i) Wl[i] = (const float*)d_in[2 + i];
  const float* Wroot = (const float*)d_in[11];
  float* out = (float*)d_out;

  float* bufA = (float*)d_ws;                                        // up to B*512*8 f32 = 16MB
  float* bufB = (float*)((char*)d_ws + (size_t)BB * VV * KK * 4);    // up to 8MB

  pc_leaf<<<dim3((BB * VV) / 256), 256, 0, stream>>>(x, W0, bufA);

  float* cur = bufA;
  float* nxt = bufB;
  int R = 256;
  for (int l = 0; l < 9; ++l) {
    pc_layer<<<dim3(R, BB / 128), 256, 0, stream>>>(cur, Wl[l], nxt, R);
    float* tmp = cur; cur = nxt; nxt = tmp;
    R >>= 1;
  }

  pc_root<<<dim3(BB / 256), 256, 0, stream>>>(cur, Wroot, out);
}